// CrossAttention_59287728554501
// MI455X (gfx1250) — compile-verified
//
#include <hip/hip_runtime.h>
#include <hip/hip_bf16.h>
#include <math.h>
#include <stdint.h>

typedef __attribute__((ext_vector_type(2))) float v2f;
typedef __attribute__((ext_vector_type(8))) float v8f;
typedef __attribute__((ext_vector_type(4))) unsigned int u32x4;
typedef __attribute__((ext_vector_type(8))) int i32x8;
typedef __attribute__((ext_vector_type(4))) int i32x4;

#define B_  2
#define C_  256
#define H_  64
#define W_  64
#define NP  (B_ * H_ * W_)      // 8192 pixels
#define A_  64
#define N_  10
#define HS  32
#define WS  32
#define MT  (HS * WS * N_)      // 10240 keys
#define HW  (H_ * W_)           // 4096
#define SNPAD 272               // 256 + 16 DWORD TDM pad -> conflict-free banks

// ---------------------------------------------------------------------------
// Prep 1: f[p][a] = sum_c features[b,c,h,w] * Wq[a,c]   (one block per pixel)
// ---------------------------------------------------------------------------
__global__ void __launch_bounds__(256) compute_f_kernel(
    const float* __restrict__ features, const float* __restrict__ Wq,
    float* __restrict__ f) {
  __shared__ float featv[C_];
  const int p = blockIdx.x;                 // 0..8191
  const int b = p >> 12;
  const int q = p & (HW - 1);
  const int c = threadIdx.x;
  featv[c] = features[b * (C_ * HW) + c * HW + q];
  __syncthreads();
  if (threadIdx.x < A_) {
    const int a = threadIdx.x;
    const float* wrow = Wq + a * C_;
    float acc = 0.0f;
    #pragma unroll 8
    for (int cc = 0; cc < C_; ++cc) acc = fmaf(featv[cc], wrow[cc], acc);
    f[p * A_ + a] = acc;
  }
}

// ---------------------------------------------------------------------------
// Prep 2: per key m=(i*WS+j)*N+n:
//   kT[m][a] = sum_c Wk[a,c]*supports[n,c,i,j]
//   sbias[m] = sum_c Ws[c]*supports[n,c,i,j]
//   sn[m][c] = supports[n,c,i,j]
// ---------------------------------------------------------------------------
__global__ void __launch_bounds__(256) compute_k_kernel(
    const float* __restrict__ supports, const float* __restrict__ Wk,
    const float* __restrict__ Wsv, float* __restrict__ kT,
    float* __restrict__ sbias, float* __restrict__ sn) {
  __shared__ float supv[C_];
  __shared__ float red[C_];
  const int m = blockIdx.x;                 // 0..10239
  const int n = m % N_;
  const int ij = m / N_;
  const int i = ij >> 5, j = ij & 31;
  const int base = n * (C_ * HS * WS) + i * WS + j;
  const int c = threadIdx.x;
  const float v = supports[base + c * (HS * WS)];
  supv[c] = v;
  sn[m * C_ + c] = v;
  red[c] = v * Wsv[c];
  for (int off = 128; off > 0; off >>= 1) {
    __syncthreads();
    if (c < off) red[c] += red[c + off];
  }
  __syncthreads();
  if (c == 0) sbias[m] = red[0];
  if (c < A_) {
    const float* wrow = Wk + c * C_;
    float acc = 0.0f;
    #pragma unroll 8
    for (int cc = 0; cc < C_; ++cc) acc = fmaf(supv[cc], wrow[cc], acc);
    kT[m * A_ + c] = acc;
  }
}

// ---------------------------------------------------------------------------
// Deterministic single-block sum (for the scalar means)
// ---------------------------------------------------------------------------
__global__ void __launch_bounds__(256) reduce_sum_kernel(
    const float* __restrict__ x, int n, float* __restrict__ out) {
  __shared__ float red[256];
  float acc = 0.0f;
  for (int idx = threadIdx.x; idx < n; idx += 256) acc += x[idx];
  red[threadIdx.x] = acc;
  for (int off = 128; off > 0; off >>= 1) {
    __syncthreads();
    if (threadIdx.x < off) red[threadIdx.x] += red[threadIdx.x + off];
  }
  __syncthreads();
  if (threadIdx.x == 0) out[0] = red[0];
}

// ---------------------------------------------------------------------------
// Subtract global scalar means in place: f -= mean(f), kT -= mean(k)
// ---------------------------------------------------------------------------
__global__ void __launch_bounds__(256) center_kernel(
    float* __restrict__ f, float* __restrict__ kT,
    const float* __restrict__ sums) {
  const float mf = sums[0] * (1.0f / (float)(NP * A_));
  const float mk = sums[1] * (1.0f / (float)(MT * A_));
  const int idx = blockIdx.x * 256 + threadIdx.x;
  if (idx < NP * A_) f[idx] -= mf;
  if (idx < MT * A_) kT[idx] -= mk;
}

// ---------------------------------------------------------------------------
// TDM: DMA one 16 x 256 f32 tile of sn into LDS, padded +16 DWORDs per
// 256-DWORD row so LDS rows land at stride SNPAD (=272) floats.
// D# per CDNA5 ISA (2-D tile, groups 2/3 zero).
// This toolchain exposes the 6-arg builtin form:
//   (u32x4 g0, i32x8 g1, i32x4, i32x4, i32x8, i32 cpol)
// ---------------------------------------------------------------------------
__device__ __forceinline__ void tdm_load_sn_tile(const float* gsrc,
                                                 uint32_t lds_addr) {
  const uint64_t ga = (uint64_t)(uintptr_t)gsrc;
  u32x4 g0;
  g0[0] = 1u;                                        // count=1, user mode
  g0[1] = lds_addr;                                  // lds_addr [63:32]
  g0[2] = (uint32_t)ga;                              // global_addr lo
  g0[3] = ((uint32_t)(ga >> 32) & 0x01FFFFFFu)       // global_addr[56:32]
          | 0x80000000u;                             // type=2 ("image")
  i32x8 g1;
  g1[0] = (int)0x1FD20000u;  // data_size=4B | pad_en | pad_int=256DW | pad_amt=16DW
  g1[1] = (int)(256u << 16); // atomic_barrier=0, tensor_dim0=256 (lo16)
  g1[2] = (int)(16u << 16);  // tensor_dim0 hi=0, tensor_dim1=16 (lo16)
  g1[3] = (int)(256u << 16); // tensor_dim1 hi=0, tile_dim0=256
  g1[4] = 16;                // tile_dim1=16, tile_dim2=0
  g1[5] = 256;               // tensor_dim0_stride=256 (lo32)
  g1[6] = 0;                 // stride hi, tensor_dim1_stride lo
  g1[7] = 0;
  const i32x4 z4 = {0, 0, 0, 0};
  const i32x8 z8 = {0, 0, 0, 0, 0, 0, 0, 0};
  __builtin_amdgcn_tensor_load_to_lds(g0, g1, z4, z4, z8, 0);
}

// ---------------------------------------------------------------------------
// Fused flash-style attention. 4 waves/block, each wave owns a 16-pixel tile;
// all waves walk the key chunks in lockstep, so the 16x256 sn (V) tile is
// fetched ONCE per block per chunk by the Tensor Data Mover into a
// double-buffered LDS tile, overlapped with the WMMA work via TENSORcnt.
// Pass 1: logits via f32 WMMA chain (K=64), track row max.
// Pass 2: recompute logits, exp, row sum, out^T += sn^T x P^T via WMMA,
//         P tile bounced through LDS for the B-fragment layout.
// Output tile is (16c x 16p) = exactly the (B,C,H,W) layout.
// ---------------------------------------------------------------------------
__global__ void __launch_bounds__(128) fused_attn_kernel(
    const float* __restrict__ f, const float* __restrict__ kT,
    const float* __restrict__ sb, const float* __restrict__ sn,
    float* __restrict__ out) {
  __shared__ float ldsSN[2][16][SNPAD];   // double-buffered V tile (TDM dest)
  __shared__ float ldsP[4][16][16];
  __shared__ float ldsSum[4][16];

  const int wave = threadIdx.x >> 5;
  const int lane = threadIdx.x & 31;
  const int half = lane >> 4;   // wave32 lane half (K split per ISA layouts)
  const int l16  = lane & 15;
  const int p0 = (blockIdx.x * 4 + wave) * 16;

  // A-fragments of the 16x64 query tile: lane<16 holds K=4t..4t+1 of row l16,
  // lane>=16 holds K=4t+2..4t+3 (ISA 16x4 f32 A layout).
  v2f afrag[16];
  {
    const float* frow = f + (p0 + l16) * A_ + half * 2;
    #pragma unroll
    for (int t = 0; t < 16; ++t)
      afrag[t] = *(const v2f*)(frow + 4 * t);
  }

  // ---- Pass 1: row maxima ------------------------------------------------
  float vmax[8];
  #pragma unroll
  for (int r = 0; r < 8; ++r) vmax[r] = -3.402823466e38f;

  for (int mc = 0; mc < MT; mc += 16) {
    v8f L = {0.f, 0.f, 0.f, 0.f, 0.f, 0.f, 0.f, 0.f};
    const float* krow = kT + (mc + l16) * A_ + half * 2;
    #pragma unroll
    for (int t = 0; t < 16; ++t) {
      v2f bfrag = *(const v2f*)(krow + 4 * t);
      L = __builtin_amdgcn_wmma_f32_16x16x4_f32(false, afrag[t], false, bfrag,
                                                (short)0, L, false, false);
    }
    const float bias = 0.1f * sb[mc + l16];
    #pragma unroll
    for (int r = 0; r < 8; ++r) vmax[r] = fmaxf(vmax[r], L[r] + bias);
  }
  #pragma unroll
  for (int r = 0; r < 8; ++r) {   // reduce across the 16-lane column group
    vmax[r] = fmaxf(vmax[r], __shfl_xor(vmax[r], 1));
    vmax[r] = fmaxf(vmax[r], __shfl_xor(vmax[r], 2));
    vmax[r] = fmaxf(vmax[r], __shfl_xor(vmax[r], 4));
    vmax[r] = fmaxf(vmax[r], __shfl_xor(vmax[r], 8));
  }

  // ---- Pass 2: exp / rowsum / out^T accumulation -------------------------
  float vsum[8];
  #pragma unroll
  for (int r = 0; r < 8; ++r) vsum[r] = 0.0f;
  v8f acc[16];
  const v8f vzero = {0.f, 0.f, 0.f, 0.f, 0.f, 0.f, 0.f, 0.f};
  #pragma unroll
  for (int ct = 0; ct < 16; ++ct) acc[ct] = vzero;

  const uint32_t ldsb0 = (uint32_t)(uintptr_t)&ldsSN[0][0][0];
  const uint32_t ldsb1 = (uint32_t)(uintptr_t)&ldsSN[1][0][0];
  if (threadIdx.x == 0) tdm_load_sn_tile(sn, ldsb0);   // prologue DMA chunk 0

  for (int mc = 0; mc < MT; mc += 16) {
    const int buf = (mc >> 4) & 1;
    if (threadIdx.x == 0) {
      if (mc + 16 < MT) {                 // prefetch next chunk, other buffer
        tdm_load_sn_tile(sn + (size_t)(mc + 16) * C_, buf ? ldsb0 : ldsb1);
        __builtin_amdgcn_s_wait_tensorcnt(1);   // current chunk landed
      } else {
        __builtin_amdgcn_s_wait_tensorcnt(0);
      }
    }
    __syncthreads();                      // V tile visible to all 4 waves

    v8f L = vzero;
    const float* krow = kT + (mc + l16) * A_ + half * 2;
    #pragma unroll
    for (int t = 0; t < 16; ++t) {
      v2f bfrag = *(const v2f*)(krow + 4 * t);
      L = __builtin_amdgcn_wmma_f32_16x16x4_f32(false, afrag[t], false, bfrag,
                                                (short)0, L, false, false);
    }
    const float bias = 0.1f * sb[mc + l16];
    #pragma unroll
    for (int r = 0; r < 8; ++r) {
      const float pv = __expf(L[r] + bias - vmax[r]);
      vsum[r] += pv;
      ldsP[wave][half * 8 + r][l16] = pv;     // C layout -> LDS p x m
    }
    __syncthreads();
    v2f bp[4];                                // B fragments of P^T (K=m, N=p)
    #pragma unroll
    for (int t = 0; t < 4; ++t)
      bp[t] = *(const v2f*)&ldsP[wave][l16][4 * t + 2 * half];
    __syncthreads();

    #pragma unroll
    for (int ct = 0; ct < 16; ++ct) {
      const int cidx = ct * 16 + l16;
      #pragma unroll
      for (int t = 0; t < 4; ++t) {
        const int mrow = 4 * t + 2 * half;
        v2f av;                               // A frag of sn^T (M=c, K=m)
        av[0] = ldsSN[buf][mrow][cidx];
        av[1] = ldsSN[buf][mrow + 1][cidx];
        acc[ct] = __builtin_amdgcn_wmma_f32_16x16x4_f32(false, av, false, bp[t],
                                                        (short)0, acc[ct],
                                                        false, false);
      }
    }
    __syncthreads();   // buffer fully consumed before its next TDM refill
  }

  #pragma unroll
  for (int r = 0; r < 8; ++r) {
    vsum[r] += __shfl_xor(vsum[r], 1);
    vsum[r] += __shfl_xor(vsum[r], 2);
    vsum[r] += __shfl_xor(vsum[r], 4);
    vsum[r] += __shfl_xor(vsum[r], 8);
  }
  if (l16 == 0) {
    #pragma unroll
    for (int r = 0; r < 8; ++r) ldsSum[wave][half * 8 + r] = vsum[r];
  }
  __syncthreads();
  const float inv = 1.0f / ldsSum[wave][l16];

  const int b = p0 >> 12;
  const int q = (p0 & (HW - 1)) + l16;
  float* outb = out + (size_t)b * C_ * HW + q;
  #pragma unroll
  for (int ct = 0; ct < 16; ++ct) {
    #pragma unroll
    for (int r = 0; r < 8; ++r) {
      const int c = ct * 16 + half * 8 + r;   // D layout: M=c0+r / c0+8+r
      outb[(size_t)c * HW] = acc[ct][r] * inv;
    }
  }
}

// ---------------------------------------------------------------------------
extern "C" void kernel_launch(void* const* d_in, const int* in_sizes, int n_in,
                              void* d_out, int out_size, void* d_ws, size_t ws_size,
                              hipStream_t stream) {
  const float* features = (const float*)d_in[0];   // (2,256,64,64)
  const float* supports = (const float*)d_in[1];   // (10,256,32,32)
  const float* Wq       = (const float*)d_in[2];   // (64,256)
  const float* Wk       = (const float*)d_in[3];   // (64,256)
  const float* Wsv      = (const float*)d_in[4];   // (1,256)
  float* out = (float*)d_out;                      // (2,256,64,64)

  float* wf   = (float*)d_ws;          // f:  8192*64
  float* wkT  = wf   + NP * A_;        // kT: 10240*64
  float* wsb  = wkT  + MT * A_;        // s:  10240
  float* wsn  = wsb  + MT;             // sn: 10240*256
  float* wsum = wsn  + (size_t)MT * C_;// 2 scalars

  compute_f_kernel<<<NP, 256, 0, stream>>>(features, Wq, wf);
  compute_k_kernel<<<MT, 256, 0, stream>>>(supports, Wk, Wsv, wkT, wsb, wsn);
  reduce_sum_kernel<<<1, 256, 0, stream>>>(wf, NP * A_, wsum);
  reduce_sum_kernel<<<1, 256, 0, stream>>>(wkT, MT * A_, wsum + 1);
  center_kernel<<<(MT * A_ + 255) / 256, 256, 0, stream>>>(wf, wkT, wsum);
  fused_attn_kernel<<<NP / 64, 128, 0, stream>>>(wf, wkT, wsb, wsn, out);
}